// VarianceReductedAggregator_28183575396850
// MI455X (gfx1250) — compile-verified
//
#include <hip/hip_runtime.h>
#include <math.h>

#define DIM 128
#define ROWS_PER_BLOCK 8   // 8 x 32-bit gather indices per TDM descriptor

typedef unsigned int v4u __attribute__((ext_vector_type(4)));
typedef int          v8i __attribute__((ext_vector_type(8)));
typedef int          v4i __attribute__((ext_vector_type(4)));

// ---------------------------------------------------------------------------
// Zero output accumulator + workspace reduction slots (out is accumulated
// into by two SpMM kernels, so it must be zeroed every call).
// ---------------------------------------------------------------------------
__global__ void vra_zero_kernel(float* __restrict__ out, double* __restrict__ acc, int n) {
  int i = blockIdx.x * blockDim.x + threadIdx.x;
  if (i < n) out[i] = 0.0f;
  if (i < 4) acc[i] = 0.0;
}

// ---------------------------------------------------------------------------
// TDM gather of old_activation[input_fields] rows into LDS (8 rows / block,
// one wave per block so every descriptor field is wave-uniform), then compute
// partial sums for sim = <inputs, old> / ||inputs|| / ||old||.
// ---------------------------------------------------------------------------
__global__ __launch_bounds__(32)
void vra_gather_sim_kernel(const float* __restrict__ inputs,
                           const float* __restrict__ old_act,
                           const int*   __restrict__ fields,
                           double*      __restrict__ acc,
                           int batch, int ndata) {
  __shared__ float smem[ROWS_PER_BLOCK * DIM];  // sole LDS object -> offset 0

  const int base  = blockIdx.x * ROWS_PER_BLOCK;
  const int nrows = (batch - base) < ROWS_PER_BLOCK ? (batch - base) : ROWS_PER_BLOCK;
  const int lane  = threadIdx.x;  // 0..31

  // Uniform (per-wave) gather indices -> SGPRs
  int idx[ROWS_PER_BLOCK];
#pragma unroll
  for (int k = 0; k < ROWS_PER_BLOCK; ++k)
    idx[k] = (k < nrows) ? fields[base + k] : 0;

#if defined(__gfx1250__)
  {
    const unsigned long long ga = (unsigned long long)(size_t)old_act;
    // D# group 0: count=1 valid, gather_mode=1, 32-bit indices, lds_addr=0,
    // global_addr = old_activation base, type=2 ("image").
    v4u g0 = { 1u | (1u << 30) | (1u << 31),
               0u,
               (unsigned)(ga & 0xFFFFFFFFu),
               (unsigned)((ga >> 32) & 0x1FFFFFFu) | (2u << 30) };
    // D# group 1: data_size=4B; tensor_dim0=DIM; tensor_dim1=ndata;
    // tile_dim0=DIM; tile_dim1=#indices; tensor_dim0_stride=DIM.
    v8i g1 = { (int)0x00020000,
               (int)((DIM & 0xFFFF) << 16),
               (int)(((unsigned)ndata & 0xFFFFu) << 16) | ((DIM >> 16) & 0xFFFF),
               (int)((((unsigned)ndata >> 16) & 0xFFFFu) | ((unsigned)DIM << 16)),
               nrows,
               DIM,
               0, 0 };
    // D# groups 2/3: eight 32-bit row indices.
    v4i g2 = { idx[0], idx[1], idx[2], idx[3] };
    v4i g3 = { idx[4], idx[5], idx[6], idx[7] };
    // Trailing (unused) address group for the 6-arg toolchain form.
    v8i g4 = { 0, 0, 0, 0, 0, 0, 0, 0 };

    __builtin_amdgcn_tensor_load_to_lds(g0, g1, g2, g3, g4, 0);
    __builtin_amdgcn_s_wait_tensorcnt(0);
    asm volatile("" ::: "memory");  // compiler fence: LDS now holds gathered rows
  }
#else
  // Host pass / non-gfx1250 fallback: manual gather.
  for (int r = 0; r < nrows; ++r) {
    const size_t src = (size_t)idx[r] * DIM;
    for (int c = lane; c < DIM; c += 32) smem[r * DIM + c] = old_act[src + c];
  }
  __syncthreads();
#endif

  // Each lane owns 4 contiguous columns; sweep the 8 gathered rows.
  volatile const float* sm = smem;
  float dot = 0.0f, nin = 0.0f, nold = 0.0f;
  for (int r = 0; r < nrows; ++r) {
    const float4 a = *(const float4*)(inputs + (size_t)(base + r) * DIM + lane * 4);
    const float o0 = sm[r * DIM + lane * 4 + 0];
    const float o1 = sm[r * DIM + lane * 4 + 1];
    const float o2 = sm[r * DIM + lane * 4 + 2];
    const float o3 = sm[r * DIM + lane * 4 + 3];
    dot  += a.x * o0 + a.y * o1 + a.z * o2 + a.w * o3;
    nin  += a.x * a.x + a.y * a.y + a.z * a.z + a.w * a.w;
    nold += o0 * o0 + o1 * o1 + o2 * o2 + o3 * o3;
  }

  // wave32 tree reduction
#pragma unroll
  for (int off = 16; off > 0; off >>= 1) {
    dot  += __shfl_down(dot,  off, 32);
    nin  += __shfl_down(nin,  off, 32);
    nold += __shfl_down(nold, off, 32);
  }
  if (lane == 0) {
    __hip_atomic_fetch_add(&acc[0], (double)dot,  __ATOMIC_RELAXED, __HIP_MEMORY_SCOPE_AGENT);
    __hip_atomic_fetch_add(&acc[1], (double)nin,  __ATOMIC_RELAXED, __HIP_MEMORY_SCOPE_AGENT);
    __hip_atomic_fetch_add(&acc[2], (double)nold, __ATOMIC_RELAXED, __HIP_MEMORY_SCOPE_AGENT);
  }
}

// ---------------------------------------------------------------------------
// out[r] += v * (inputs[c] - old_activation[fields[c]])   (subsampled support)
// One wave per edge; lane handles 4 floats (32 lanes * 4 = 128 = DIM).
// ---------------------------------------------------------------------------
__global__ __launch_bounds__(256)
void vra_spmm_sub_kernel(const int* __restrict__ rows, const int* __restrict__ cols,
                         const float* __restrict__ vals,
                         const float* __restrict__ inputs,
                         const float* __restrict__ old_act,
                         const int* __restrict__ fields,
                         float* __restrict__ out, int nnz) {
  const int lane = threadIdx.x & 31;
  const int e    = blockIdx.x * 8 + (threadIdx.x >> 5);
  if (e >= nnz) return;
  const int   r = rows[e];
  const int   c = cols[e];
  const float v = vals[e];
  const int   f = fields[c];
  const float4 a = *(const float4*)(inputs  + (size_t)c * DIM + lane * 4);
  const float4 b = *(const float4*)(old_act + (size_t)f * DIM + lane * 4);
  float* o = out + (size_t)r * DIM + lane * 4;
  __hip_atomic_fetch_add(o + 0, v * (a.x - b.x), __ATOMIC_RELAXED, __HIP_MEMORY_SCOPE_AGENT);
  __hip_atomic_fetch_add(o + 1, v * (a.y - b.y), __ATOMIC_RELAXED, __HIP_MEMORY_SCOPE_AGENT);
  __hip_atomic_fetch_add(o + 2, v * (a.z - b.z), __ATOMIC_RELAXED, __HIP_MEMORY_SCOPE_AGENT);
  __hip_atomic_fetch_add(o + 3, v * (a.w - b.w), __ATOMIC_RELAXED, __HIP_MEMORY_SCOPE_AGENT);
}

// ---------------------------------------------------------------------------
// out[r] += v * old_activation[c]   (full support; rows are L2-resident)
// ---------------------------------------------------------------------------
__global__ __launch_bounds__(256)
void vra_spmm_sup_kernel(const int* __restrict__ rows, const int* __restrict__ cols,
                         const float* __restrict__ vals,
                         const float* __restrict__ old_act,
                         float* __restrict__ out, int nnz) {
  const int lane = threadIdx.x & 31;
  const int e    = blockIdx.x * 8 + (threadIdx.x >> 5);
  if (e >= nnz) return;
  const int   r = rows[e];
  const int   c = cols[e];
  const float v = vals[e];
  const float4 b = *(const float4*)(old_act + (size_t)c * DIM + lane * 4);
  float* o = out + (size_t)r * DIM + lane * 4;
  __hip_atomic_fetch_add(o + 0, v * b.x, __ATOMIC_RELAXED, __HIP_MEMORY_SCOPE_AGENT);
  __hip_atomic_fetch_add(o + 1, v * b.y, __ATOMIC_RELAXED, __HIP_MEMORY_SCOPE_AGENT);
  __hip_atomic_fetch_add(o + 2, v * b.z, __ATOMIC_RELAXED, __HIP_MEMORY_SCOPE_AGENT);
  __hip_atomic_fetch_add(o + 3, v * b.w, __ATOMIC_RELAXED, __HIP_MEMORY_SCOPE_AGENT);
}

// ---------------------------------------------------------------------------
// sim = dot / sqrt(|inputs|^2) / sqrt(|old|^2)
// ---------------------------------------------------------------------------
__global__ void vra_finalize_kernel(const double* __restrict__ acc, float* __restrict__ sim_out) {
  if (threadIdx.x == 0 && blockIdx.x == 0) {
    const double s = acc[0] / sqrt(acc[1]) / sqrt(acc[2]);
    *sim_out = (float)s;
  }
}

extern "C" void kernel_launch(void* const* d_in, const int* in_sizes, int n_in,
                              void* d_out, int out_size, void* d_ws, size_t ws_size,
                              hipStream_t stream) {
  const float* inputs   = (const float*)d_in[0];
  const float* old_act  = (const float*)d_in[1];
  const int*   fields   = (const int*)d_in[2];
  const int*   sub_rows = (const int*)d_in[3];
  const int*   sub_cols = (const int*)d_in[4];
  const float* sub_vals = (const float*)d_in[5];
  const int*   sup_rows = (const int*)d_in[6];
  const int*   sup_cols = (const int*)d_in[7];
  const float* sup_vals = (const float*)d_in[8];

  const int BD    = in_sizes[0];          // BATCH * DIM
  const int B     = in_sizes[2];          // BATCH
  const int N     = in_sizes[1] / DIM;    // N_DATA
  const int e_sub = in_sizes[3];
  const int e_sup = in_sizes[6];

  float*  out = (float*)d_out;
  double* acc = (double*)d_ws;            // 3 f64 reduction slots

  vra_zero_kernel<<<(BD + 255) / 256, 256, 0, stream>>>(out, acc, BD);

  vra_gather_sim_kernel<<<(B + ROWS_PER_BLOCK - 1) / ROWS_PER_BLOCK, 32, 0, stream>>>(
      inputs, old_act, fields, acc, B, N);

  vra_spmm_sub_kernel<<<(e_sub + 7) / 8, 256, 0, stream>>>(
      sub_rows, sub_cols, sub_vals, inputs, old_act, fields, out, e_sub);

  vra_spmm_sup_kernel<<<(e_sup + 7) / 8, 256, 0, stream>>>(
      sup_rows, sup_cols, sup_vals, old_act, out, e_sup);

  vra_finalize_kernel<<<1, 32, 0, stream>>>(acc, out + BD);
}